// SocialRecommender_87866440942272
// MI455X (gfx1250) — compile-verified
//
#include <hip/hip_runtime.h>

#define EMB 64

typedef float v2f __attribute__((ext_vector_type(2)));
typedef float v8f __attribute__((ext_vector_type(8)));

// ---------------------------------------------------------------------------
// Hardware f32 atomic add (global_atomic_add_f32, no CAS loop).
// ---------------------------------------------------------------------------
__device__ __forceinline__ void atomAddF32(float* p, float v) {
    unsafeAtomicAdd(p, v);
}

// ---------------------------------------------------------------------------
// Dense helpers (float4 vectorized, grid-stride)
// ---------------------------------------------------------------------------
__global__ void k_copy(const float* __restrict__ in, float* __restrict__ out, int n4) {
    int i = blockIdx.x * blockDim.x + threadIdx.x;
    int stride = gridDim.x * blockDim.x;
    const float4* i4 = (const float4*)in;
    float4* o4 = (float4*)out;
    for (; i < n4; i += stride) o4[i] = i4[i];
}

__global__ void k_zero(float* __restrict__ out, int n4) {
    int i = blockIdx.x * blockDim.x + threadIdx.x;
    int stride = gridDim.x * blockDim.x;
    float4* o4 = (float4*)out;
    float4 z = make_float4(0.f, 0.f, 0.f, 0.f);
    for (; i < n4; i += stride) o4[i] = z;
}

__global__ void k_add(const float* __restrict__ in, float* __restrict__ acc, int n4) {
    int i = blockIdx.x * blockDim.x + threadIdx.x;
    int stride = gridDim.x * blockDim.x;
    const float4* i4 = (const float4*)in;
    float4* a4 = (float4*)acc;
    for (; i < n4; i += stride) {
        float4 a = a4[i], b = i4[i];
        a.x += b.x; a.y += b.y; a.z += b.z; a.w += b.w;
        a4[i] = a;
    }
}

// ---------------------------------------------------------------------------
// COO SpMM, both directions from one edge read:
//   out_row[r][:] += v * xcol[c][:]     (R  @ X_item)
//   out_col[c][:] += v * xrow[r][:]     (R^T @ X_user)
// 16 lanes per nonzero, float4 per lane (16*4 = 64 dims).
// Gathers hit the 192MB L2 (tables are 25.6/12.8 MB); scatters use native
// f32 atomics. Index streams prefetched ahead (global_prefetch_b8).
// ---------------------------------------------------------------------------
__global__ void k_spmm_dual(const int* __restrict__ rows, const int* __restrict__ cols,
                            const float* __restrict__ vals,
                            const float* __restrict__ xcol, const float* __restrict__ xrow,
                            float* __restrict__ out_row, float* __restrict__ out_col,
                            int nnz) {
    long gid = (long)blockIdx.x * blockDim.x + threadIdx.x;
    int e   = (int)(gid >> 4);
    int sub = (int)(gid & 15);
    if (e >= nnz) return;
    if (sub == 0) {
        int pe = e + 4096; if (pe >= nnz) pe = nnz - 1;
        __builtin_prefetch(rows + pe, 0, 1);
        __builtin_prefetch(cols + pe, 0, 1);
        __builtin_prefetch(vals + pe, 0, 1);
    }
    int r = rows[e];
    int c = cols[e];
    float v = vals[e];
    float4 xc = ((const float4*)(xcol + (long)c * EMB))[sub];
    float4 xr = ((const float4*)(xrow + (long)r * EMB))[sub];
    float* pr = out_row + (long)r * EMB + sub * 4;
    float* pc = out_col + (long)c * EMB + sub * 4;
    atomAddF32(pr + 0, v * xc.x); atomAddF32(pr + 1, v * xc.y);
    atomAddF32(pr + 2, v * xc.z); atomAddF32(pr + 3, v * xc.w);
    atomAddF32(pc + 0, v * xr.x); atomAddF32(pc + 1, v * xr.y);
    atomAddF32(pc + 2, v * xr.z); atomAddF32(pc + 3, v * xr.w);
}

// Single-direction SpMM (social graph): out[r][:] += v * x[c][:]
__global__ void k_spmm_single(const int* __restrict__ rows, const int* __restrict__ cols,
                              const float* __restrict__ vals,
                              const float* __restrict__ x, float* __restrict__ out,
                              int nnz) {
    long gid = (long)blockIdx.x * blockDim.x + threadIdx.x;
    int e   = (int)(gid >> 4);
    int sub = (int)(gid & 15);
    if (e >= nnz) return;
    if (sub == 0) {
        int pe = e + 4096; if (pe >= nnz) pe = nnz - 1;
        __builtin_prefetch(rows + pe, 0, 1);
        __builtin_prefetch(cols + pe, 0, 1);
        __builtin_prefetch(vals + pe, 0, 1);
    }
    int r = rows[e];
    int c = cols[e];
    float v = vals[e];
    float4 xc = ((const float4*)(x + (long)c * EMB))[sub];
    float* pr = out + (long)r * EMB + sub * 4;
    atomAddF32(pr + 0, v * xc.x); atomAddF32(pr + 1, v * xc.y);
    atomAddF32(pr + 2, v * xc.z); atomAddF32(pr + 3, v * xc.w);
}

// ---------------------------------------------------------------------------
// Finalize: comb = wa*pa + wb*pb per element, then L2-normalize each row.
// One wave handles 16 rows. Cross-lane sum-of-squares reduction done with
// V_WMMA_F32_16X16X4_F32:
//   A (16x4 f32): lane L<16 owns {A[L,0],A[L,1]}, lane L>=16 owns {A[L-16,2],A[L-16,3]}
//   -> lane places partial sum-of-squares of its 32 assigned elements there.
//   B = ones(4x16)  ->  D[m,n] = sum_k A[m,k] = rowwise sum of squares (all n equal).
// EXEC is all-ones inside (guard is wave-uniform; row counts divide by 16).
// ---------------------------------------------------------------------------
__global__ void k_finalize(const float* __restrict__ pa, const float* __restrict__ pb,
                           float wa, float wb, float* __restrict__ out, int nrows) {
    int lane = threadIdx.x & 31;
    int wave = (int)(((long)blockIdx.x * blockDim.x + threadIdx.x) >> 5);
    int base = wave * 16;
    if (base >= nrows) return;          // wave-uniform exit (nrows % 16 == 0)

    int r    = lane & 15;               // row within the 16-row tile
    int half = lane >> 4;               // which pair of columns per chunk
    long rowoff = (long)(base + r) * EMB;
    const float2* a2 = (const float2*)(pa + rowoff);
    const float2* b2 = (const float2*)(pb + rowoff);

    float valbuf[32];
    float s0 = 0.f, s1 = 0.f;
#pragma unroll
    for (int t = 0; t < 16; ++t) {
        int idx = 2 * t + half;         // float2 index => columns 4t+2*half, +1
        float2 ca = a2[idx];
        float2 cb = b2[idx];
        float c0 = wa * ca.x + wb * cb.x;
        float c1 = wa * ca.y + wb * cb.y;
        valbuf[2 * t]     = c0;
        valbuf[2 * t + 1] = c1;
        s0 += c0 * c0;
        s1 += c1 * c1;
    }

    v2f a;     a.x = s0;  a.y = s1;
    v2f bones; bones.x = 1.f; bones.y = 1.f;
    v8f cz = {};
    v8f d = __builtin_amdgcn_wmma_f32_16x16x4_f32(
        /*neg_a=*/false, a, /*neg_b=*/false, bones,
        /*c_mod=*/(short)0, cz, /*reuse_a=*/false, /*reuse_b=*/false);

    // D layout: lanes 0-15 hold rows 0-7 in VGPRs 0-7; lanes 16-31 hold rows 8-15.
    // All N columns are identical, so any lane in the right half has our row's value.
    float myv = 0.f;
#pragma unroll
    for (int j = 0; j < 8; ++j)
        if ((r & 7) == j) myv = d[j];
    int src = (r < 8) ? r : (r + 16);
    float sumsq = __shfl(myv, src, 32);

    float norm  = sqrtf(sumsq);
    float scale = 1.f / fmaxf(norm, 1e-12f);

    float2* o2 = (float2*)(out + rowoff);
#pragma unroll
    for (int t = 0; t < 16; ++t) {
        int idx = 2 * t + half;
        float2 w;
        w.x = valbuf[2 * t] * scale;
        w.y = valbuf[2 * t + 1] * scale;
        o2[idx] = w;
    }
}

// ---------------------------------------------------------------------------
// Host orchestration
// ---------------------------------------------------------------------------
extern "C" void kernel_launch(void* const* d_in, const int* in_sizes, int n_in,
                              void* d_out, int out_size, void* d_ws, size_t ws_size,
                              hipStream_t stream) {
    const float* user_emb = (const float*)d_in[0];
    const float* item_emb = (const float*)d_in[1];
    const float* r_vals   = (const float*)d_in[2];
    const float* s_vals   = (const float*)d_in[3];
    const int*   r_rows   = (const int*)d_in[4];
    const int*   r_cols   = (const int*)d_in[5];
    const int*   s_rows   = (const int*)d_in[6];
    const int*   s_cols   = (const int*)d_in[7];

    const int U = in_sizes[0] / EMB;     // 100000
    const int I = in_sizes[1] / EMB;     // 50000
    const int nnz_r = in_sizes[2];       // 3.2M
    const int nnz_s = in_sizes[3];       // 2.0M
    const long U64 = (long)U * EMB;
    const long I64 = (long)I * EMB;
    const int nU4 = (int)(U64 / 4);
    const int nI4 = (int)(I64 / 4);

    float* outU = (float*)d_out;         // doubles as u_sum accumulator
    float* outI = outU + U64;            // doubles as i_sum accumulator

    float* A = (float*)d_ws;             // user cur   (U*64)
    float* B = A + U64;                  // user next  (U*64)
    float* C = B + U64;                  // item cur   (I*64)
    float* D = C + I64;                  // item next  (I*64)
    float* E = D + I64;                  // social sum (U*64)

    const int TB = 256;
    const int DENSE_BLK = 1024;          // grid-stride dense kernels

    // init: cur = emb, sums = emb
    k_copy<<<DENSE_BLK, TB, 0, stream>>>(user_emb, A, nU4);
    k_copy<<<DENSE_BLK, TB, 0, stream>>>(item_emb, C, nI4);
    k_copy<<<DENSE_BLK, TB, 0, stream>>>(user_emb, outU, nU4);
    k_copy<<<DENSE_BLK, TB, 0, stream>>>(item_emb, outI, nI4);

    // bipartite propagation: 3 layers
    long rthreads = (long)nnz_r * 16;
    int rblocks = (int)((rthreads + TB - 1) / TB);
    for (int l = 0; l < 3; ++l) {
        k_zero<<<DENSE_BLK, TB, 0, stream>>>(B, nU4);
        k_zero<<<DENSE_BLK, TB, 0, stream>>>(D, nI4);
        k_spmm_dual<<<rblocks, TB, 0, stream>>>(r_rows, r_cols, r_vals,
                                                C /*items by col*/, A /*users by row*/,
                                                B /*u_next*/, D /*i_next*/, nnz_r);
        k_add<<<DENSE_BLK, TB, 0, stream>>>(B, outU, nU4);
        k_add<<<DENSE_BLK, TB, 0, stream>>>(D, outI, nI4);
        { float* t = A; A = B; B = t; }
        { float* t = C; C = D; D = t; }
    }

    // social propagation: 2 layers (reuse A/B ping-pong)
    k_copy<<<DENSE_BLK, TB, 0, stream>>>(user_emb, A, nU4);
    k_copy<<<DENSE_BLK, TB, 0, stream>>>(user_emb, E, nU4);
    long sthreads = (long)nnz_s * 16;
    int sblocks = (int)((sthreads + TB - 1) / TB);
    for (int l = 0; l < 2; ++l) {
        k_zero<<<DENSE_BLK, TB, 0, stream>>>(B, nU4);
        k_spmm_single<<<sblocks, TB, 0, stream>>>(s_rows, s_cols, s_vals, A, B, nnz_s);
        k_add<<<DENSE_BLK, TB, 0, stream>>>(B, E, nU4);
        { float* t = A; A = B; B = t; }
    }

    // finalize: user = normalize(0.6*u_sum/4 + 0.4*s_sum/3); item = normalize(i_sum/4)
    {
        int waves = (U + 15) / 16;
        int blocks = (waves * 32 + TB - 1) / TB;
        k_finalize<<<blocks, TB, 0, stream>>>(outU, E, 0.6f / 4.f, 0.4f / 3.f, outU, U);
    }
    {
        int waves = (I + 15) / 16;
        int blocks = (waves * 32 + TB - 1) / TB;
        k_finalize<<<blocks, TB, 0, stream>>>(outI, outI, 0.25f, 0.0f, outI, I);
    }
}